// GraphConv_12120397709966
// MI455X (gfx1250) — compile-verified
//
#include <hip/hip_runtime.h>

typedef __attribute__((ext_vector_type(2))) float v2f;
typedef __attribute__((ext_vector_type(8))) float v8f;

#define GC_N_NODES 50000
#define GC_N_EDGES 800000
#define GC_D 128   // D_IN == D_OUT == 128

// ---------------------------------------------------------------------------
// Kernel 0: zero the agg workspace (harness poisons d_ws; must re-init per call)
// ---------------------------------------------------------------------------
__global__ void gc_zero_kernel(float4* __restrict__ agg, int n4) {
    int i = blockIdx.x * blockDim.x + threadIdx.x;
    if (i < n4) agg[i] = make_float4(0.f, 0.f, 0.f, 0.f);
}

// ---------------------------------------------------------------------------
// Kernel 1: edge scatter.  One wave32 per edge; lane l handles features
// [4l, 4l+4): coalesced float4 load of x[col[e]] (32 lanes * 16B = 512B row),
// scaled by edge weight, native f32 atomics into agg[row[e]].
// ---------------------------------------------------------------------------
__global__ void gc_scatter_kernel(const float* __restrict__ x,
                                  const int*   __restrict__ erow,
                                  const int*   __restrict__ ecol,
                                  const float* __restrict__ ew,
                                  float*       __restrict__ agg) {
    unsigned gtid = blockIdx.x * blockDim.x + threadIdx.x;
    unsigned e    = gtid >> 5;          // wave index == edge index
    unsigned lane = threadIdx.x & 31u;
    if (e >= GC_N_EDGES) return;

    int   r = erow[e];
    int   c = ecol[e];
    float w = ew[e];

    const float4* xr = (const float4*)(x + (size_t)c * GC_D);
    float4 v = xr[lane];

    float* dst = agg + (size_t)r * GC_D + lane * 4u;
    unsafeAtomicAdd(dst + 0, w * v.x);   // global_atomic_add_f32
    unsafeAtomicAdd(dst + 1, w * v.y);
    unsafeAtomicAdd(dst + 2, w * v.z);
    unsafeAtomicAdd(dst + 3, w * v.w);
}

// ---------------------------------------------------------------------------
// Kernel 2: out = agg @ W^T + b via V_WMMA_F32_16X16X4_F32.
// Block = 256 threads = 8 waves; block -> one 16-row M tile, wave w -> N tile
// [16w, 16w+16).  K=128 consumed 4 at a time (32 WMMA per wave).
//
// f32 A (16x4) lane layout : lanes 0-15 row M=l, {K, K+1}; lanes 16-31 row M=l-16,
//                            {K+2, K+3}  -> per-lane float2 at (row*D + k + 2*(l>>4)).
// f32 B (4x16) lane layout : symmetric; B[k][n] = W[n][k] (W is row-major [out,in]),
//                            per-lane float2 at (W + n*D + k + 2*(l>>4)).
// f32 C/D (16x16)          : VGPR r -> row m = r + 8*(l>>4), col n = l&15.
// ---------------------------------------------------------------------------
__global__ void gc_gemm_wmma_kernel(const float* __restrict__ agg,
                                    const float* __restrict__ W,
                                    const float* __restrict__ bias,
                                    float*       __restrict__ out) {
    const int mtile = blockIdx.x;            // 0 .. 3124 (50000/16 exact)
    const int wv    = threadIdx.x >> 5;      // 0 .. 7  -> N tile
    const int lane  = threadIdx.x & 31;
    const int m0    = mtile * 16;
    const int n0    = wv * 16;
    const int half  = lane >> 4;             // 0 or 1
    const int l15   = lane & 15;

    const float* arow = agg + (size_t)(m0 + l15) * GC_D + 2 * half;
    const float* brow = W   + (size_t)(n0 + l15) * GC_D + 2 * half;

    v8f c = {0.f, 0.f, 0.f, 0.f, 0.f, 0.f, 0.f, 0.f};

#pragma unroll 8
    for (int k = 0; k < GC_D; k += 4) {
        v2f a = *(const v2f*)(arow + k);     // 8B-aligned global_load_b64
        v2f b = *(const v2f*)(brow + k);
        c = __builtin_amdgcn_wmma_f32_16x16x4_f32(
                /*neg_a=*/false, a, /*neg_b=*/false, b,
                /*c_mod=*/(short)0, c, /*reuse_a=*/false, /*reuse_b=*/false);
    }

    const float bb = bias[n0 + l15];
    float* obase = out + (size_t)(m0 + half * 8) * GC_D + (n0 + l15);
#pragma unroll
    for (int r = 0; r < 8; ++r) {
        obase[(size_t)r * GC_D] = c[r] + bb;
    }
}

// ---------------------------------------------------------------------------
extern "C" void kernel_launch(void* const* d_in, const int* in_sizes, int n_in,
                              void* d_out, int out_size, void* d_ws, size_t ws_size,
                              hipStream_t stream) {
    const float* x    = (const float*)d_in[0];
    const int*   erow = (const int*)  d_in[1];
    const int*   ecol = (const int*)  d_in[2];
    const float* ew   = (const float*)d_in[3];
    const float* W    = (const float*)d_in[4];
    const float* b    = (const float*)d_in[5];
    float*       out  = (float*)d_out;
    float*       agg  = (float*)d_ws;        // 50000*128*4 = 25.6 MB scratch

    // 1) zero agg
    const int n4 = GC_N_NODES * GC_D / 4;    // 1.6M float4
    gc_zero_kernel<<<(n4 + 255) / 256, 256, 0, stream>>>((float4*)agg, n4);

    // 2) edge scatter: one wave32 per edge -> 800k waves -> 100k blocks of 8 waves
    const long long threads = (long long)GC_N_EDGES * 32;
    gc_scatter_kernel<<<(unsigned)((threads + 255) / 256), 256, 0, stream>>>(
        x, erow, ecol, ew, agg);

    // 3) WMMA GEMM: one block per 16-row tile (3125 blocks), 8 waves cover 8 N-tiles
    gc_gemm_wmma_kernel<<<GC_N_NODES / 16, 256, 0, stream>>>(agg, W, b, out);
}